// Memory_17497696764698
// MI455X (gfx1250) — compile-verified
//
#include <hip/hip_runtime.h>
#include <math.h>

typedef float v2f __attribute__((ext_vector_type(2)));
typedef float v4f __attribute__((ext_vector_type(4)));
typedef float v8f __attribute__((ext_vector_type(8)));

#define EPSV 1e-12f

// ---------------------------------------------------------------------------
// GEMM: Y[b,n] = act( sum_c A[b,c] * W[n,c] + bias[n] )
// A: [64,1024] row-major, W: [N,1024] row-major. fp32 WMMA 16x16x4.
// One wave per 16x16 output tile; grid sized exactly (EXEC all-ones).
// ---------------------------------------------------------------------------
__global__ void gemm_ctrl_wmma(const float* __restrict__ A,
                               const float* __restrict__ W,
                               const float* __restrict__ bias,
                               float* __restrict__ Y,
                               int N, int act) {
    const int C = 1024;
    int wave  = (blockIdx.x * blockDim.x + threadIdx.x) >> 5;
    int lane  = threadIdx.x & 31;
    int tilesN = N >> 4;
    int tm = wave / tilesN;          // 0..3  (64 rows / 16)
    int tn = wave % tilesN;

    int arow  = tm * 16 + (lane & 15);
    int bcol  = tn * 16 + (lane & 15);
    int khalf = (lane >> 4) << 1;    // 0 or 2

    const float* aptr = A + arow * C + khalf;
    const float* bptr = W + bcol * C + khalf;

    v8f c = {};
    #pragma unroll 8
    for (int k = 0; k < C; k += 4) {
        v2f a = *(const v2f*)(aptr + k);
        v2f b = *(const v2f*)(bptr + k);
        c = __builtin_amdgcn_wmma_f32_16x16x4_f32(false, a, false, b,
                                                  (short)0, c, false, false);
    }

    float bv   = bias[bcol];
    int   rbase = tm * 16 + ((lane >> 4) << 3);
    #pragma unroll
    for (int i = 0; i < 8; ++i) {
        float v = c[i] + bv;
        if (act == 1) v = 1.0f / (1.0f + expf(-v));   // sigmoid
        Y[(rbase + i) * N + bcol] = v;
    }
}

__device__ __forceinline__ float wave_sum(float v) {
    #pragma unroll
    for (int off = 16; off > 0; off >>= 1) v += __shfl_xor(v, off);
    return v;
}

// ---------------------------------------------------------------------------
// mix (softmax over 2) and sharp (softplus+1) per (k,b). One wave each.
// ---------------------------------------------------------------------------
__global__ void mixsharp_kernel(const float* __restrict__ ctrl,
                                const float* __restrict__ Wmix,
                                const float* __restrict__ bmix,
                                const float* __restrict__ Wsharp,
                                const float* __restrict__ bsharp,
                                float* __restrict__ mix,   // [3][64][2]
                                float* __restrict__ sharp) // [3][64]
{
    int wave = (blockIdx.x * blockDim.x + threadIdx.x) >> 5;
    int lane = threadIdx.x & 31;
    int k = wave >> 6, b = wave & 63;
    const float* c  = ctrl + b * 1024;
    const float* w0 = Wmix + (k * 2 + 0) * 1024;
    const float* w1 = Wmix + (k * 2 + 1) * 1024;
    const float* w2 = Wsharp + k * 1024;
    float s0 = 0.f, s1 = 0.f, s2 = 0.f;
    for (int i = lane; i < 1024; i += 32) {
        float cv = c[i];
        s0 += cv * w0[i]; s1 += cv * w1[i]; s2 += cv * w2[i];
    }
    s0 = wave_sum(s0); s1 = wave_sum(s1); s2 = wave_sum(s2);
    if (lane == 0) {
        float l0 = s0 + bmix[k * 2 + 0], l1 = s1 + bmix[k * 2 + 1];
        float mx = fmaxf(l0, l1);
        float e0 = expf(l0 - mx), e1 = expf(l1 - mx);
        float inv = 1.0f / (e0 + e1);
        mix[(k * 64 + b) * 2 + 0] = e0 * inv;
        mix[(k * 64 + b) * 2 + 1] = e1 * inv;
        float x  = s2 + bsharp[k];
        float sp = (x > 20.0f) ? x : log1pf(expf(x));
        sharp[k * 64 + b] = sp + 1.0f;
    }
}

// ---------------------------------------------------------------------------
// vk[b, k*512+o] = l2norm(keys[k,b,:])[o] * subs[k,b,o].  One wave per (k,b).
// ---------------------------------------------------------------------------
__global__ void vk_kernel(const float* __restrict__ keys,  // [64][1536]
                          const float* __restrict__ subs,  // [64][1536]
                          float* __restrict__ vk)          // [64][1536]
{
    int wave = (blockIdx.x * blockDim.x + threadIdx.x) >> 5;
    int lane = threadIdx.x & 31;
    int k = wave >> 6, b = wave & 63;
    const float* kp = keys + b * 1536 + k * 512;
    const float* sp = subs + b * 1536 + k * 512;
    float*       vp = vk   + b * 1536 + k * 512;
    float sq = 0.f;
    #pragma unroll
    for (int j = 0; j < 4; ++j) {
        v4f v = *(const v4f*)(kp + (lane + j * 32) * 4);
        sq += v.x * v.x + v.y * v.y + v.z * v.z + v.w * v.w;
    }
    sq = wave_sum(sq);
    float inv = 1.0f / fmaxf(sqrtf(sq), EPSV);
    #pragma unroll
    for (int j = 0; j < 4; ++j) {
        int i = (lane + j * 32) * 4;
        v4f kv = *(const v4f*)(kp + i);
        v4f sv = *(const v4f*)(sp + i);
        *(v4f*)(vp + i) = kv * inv * sv;
    }
}

// ---------------------------------------------------------------------------
// Pass 1 over memory: per (b,m) row compute ||mem+EPS|| and dots with vk[k].
// One wave per row; b128 loads (lane -> 16 contiguous bytes, coalesced).
// ---------------------------------------------------------------------------
__global__ void content_kernel(const float* __restrict__ memory, // [64][1024][512]
                               const float* __restrict__ vk,     // [64][1536]
                               float* __restrict__ content)      // [3][64][1024]
{
    int wave = (blockIdx.x * blockDim.x + threadIdx.x) >> 5;
    int lane = threadIdx.x & 31;
    int b = wave >> 10, m = wave & 1023;
    const float* row = memory + ((size_t)(b * 1024 + m)) * 512;
    const float* v0 = vk + b * 1536;
    const float* v1 = v0 + 512;
    const float* v2 = v0 + 1024;
    float sq = 0.f, d0 = 0.f, d1 = 0.f, d2 = 0.f;
    #pragma unroll
    for (int j = 0; j < 4; ++j) {
        int i = (lane + j * 32) * 4;
        v4f x  = *(const v4f*)(row + i) + EPSV;
        v4f a0 = *(const v4f*)(v0 + i);
        v4f a1 = *(const v4f*)(v1 + i);
        v4f a2 = *(const v4f*)(v2 + i);
        sq += x.x * x.x  + x.y * x.y  + x.z * x.z  + x.w * x.w;
        d0 += x.x * a0.x + x.y * a0.y + x.z * a0.z + x.w * a0.w;
        d1 += x.x * a1.x + x.y * a1.y + x.z * a1.z + x.w * a1.w;
        d2 += x.x * a2.x + x.y * a2.y + x.z * a2.z + x.w * a2.w;
    }
    sq = wave_sum(sq); d0 = wave_sum(d0); d1 = wave_sum(d1); d2 = wave_sum(d2);
    if (lane == 0) {
        float inv = 1.0f / fmaxf(sqrtf(sq), EPSV);
        content[(0 * 64 + b) * 1024 + m] = d0 * inv;
        content[(1 * 64 + b) * 1024 + m] = d1 * inv;
        content[(2 * 64 + b) * 1024 + m] = d2 * inv;
    }
}

// ---------------------------------------------------------------------------
// Addressing: softmax(loc) over M, mix with content, sharpen pow, L1 norm.
// One block (256 threads, 4 elems each) per (k,b).
// ---------------------------------------------------------------------------
__global__ void addr_kernel(const float* __restrict__ locl,    // [64][3072]
                            const float* __restrict__ content, // [3][64][1024]
                            const float* __restrict__ mix,     // [3][64][2]
                            const float* __restrict__ sharp,   // [3][64]
                            float* __restrict__ wts)           // [3][64][1024]
{
    int k = blockIdx.x >> 6, b = blockIdx.x & 63;
    int t = threadIdx.x;
    const float* ll = locl + b * 3072 + k * 1024;
    const float* ct = content + (k * 64 + b) * 1024;
    float m0 = mix[(k * 64 + b) * 2 + 0];
    float m1 = mix[(k * 64 + b) * 2 + 1];
    float s  = sharp[k * 64 + b];

    __shared__ float red[256];

    float lv[4], lmax = -INFINITY;
    #pragma unroll
    for (int j = 0; j < 4; ++j) { lv[j] = ll[t + j * 256]; lmax = fmaxf(lmax, lv[j]); }
    red[t] = lmax; __syncthreads();
    for (int off = 128; off > 0; off >>= 1) {
        if (t < off) red[t] = fmaxf(red[t], red[t + off]);
        __syncthreads();
    }
    lmax = red[0]; __syncthreads();

    float ev[4], esum = 0.f;
    #pragma unroll
    for (int j = 0; j < 4; ++j) { ev[j] = expf(lv[j] - lmax); esum += ev[j]; }
    red[t] = esum; __syncthreads();
    for (int off = 128; off > 0; off >>= 1) {
        if (t < off) red[t] += red[t + off];
        __syncthreads();
    }
    float pinv = 1.0f / red[0]; __syncthreads();

    float pw[4], psum = 0.f;
    #pragma unroll
    for (int j = 0; j < 4; ++j) {
        float mixed = ct[t + j * 256] * m0 + ev[j] * pinv * m1;
        float p = powf(mixed + EPSV, s);
        pw[j] = p; psum += fabsf(p);
    }
    red[t] = psum; __syncthreads();
    for (int off = 128; off > 0; off >>= 1) {
        if (t < off) red[t] += red[t + off];
        __syncthreads();
    }
    float winv = 1.0f / fmaxf(red[0], EPSV);
    #pragma unroll
    for (int j = 0; j < 4; ++j)
        wts[(k * 64 + b) * 1024 + t + j * 256] = pw[j] * winv;
}

// ---------------------------------------------------------------------------
// Pass 2: mem_out = memory*(1 - we*ge) + ww*obj*gw; accumulate read partials.
// Grid: 64 b x 8 chunks (128 rows each); 256 threads.
// Threads 0..127 own all 512 columns as float4 on even rows; threads
// 128..255 the same columns on odd rows -> b128 loads/stores, coalesced.
// mem_out stores are non-temporal so the 134MB write stream does not evict
// `memory` (which fits in the 192MB L2 after pass 1) from L2.
// ---------------------------------------------------------------------------
__global__ void update_kernel(const float* __restrict__ memory,
                              const float* __restrict__ obj,   // [64][512]
                              const float* __restrict__ gates, // [64][1536]
                              const float* __restrict__ wts,   // [3][64][1024]
                              float* __restrict__ mem_out,
                              float* __restrict__ partial)     // [16][64][512]
{
    int b = blockIdx.x >> 3;
    int chunk = blockIdx.x & 7;
    int t  = threadIdx.x;
    int o  = (t & 127) * 4;   // column group (float4)
    int r0 = t >> 7;          // row parity: 0 or 1

    const float* g = gates + b * 1536;
    v4f ge = *(const v4f*)(g + 512 + o);
    v4f wo = *(const v4f*)(obj + b * 512 + o) * *(const v4f*)(g + 1024 + o);

    const float* wr = wts + (0 * 64 + b) * 1024;
    const float* we = wts + (1 * 64 + b) * 1024;
    const float* ww = wts + (2 * 64 + b) * 1024;

    int m0 = chunk * 128;
    const float* mrow = memory  + ((size_t)(b * 1024 + m0)) * 512;
    float*       orow = mem_out + ((size_t)(b * 1024 + m0)) * 512;

    v4f acc = {};
    for (int mm = r0; mm < 128; mm += 2) {
        float wrv = wr[m0 + mm], wev = we[m0 + mm], wwv = ww[m0 + mm];
        v4f v = *(const v4f*)(mrow + (size_t)mm * 512 + o);
        v4f n = v - v * (ge * wev) + wo * wwv;
        __builtin_nontemporal_store(n, (v4f*)(orow + (size_t)mm * 512 + o));
        acc += n * wrv;
    }
    *(v4f*)(partial + ((size_t)(chunk * 2 + r0) * 64 + b) * 512 + o) = acc;
}

// ---------------------------------------------------------------------------
// Finalize read vector: sum 16 chunk partials, multiply by read gate.
// ---------------------------------------------------------------------------
__global__ void finalize_kernel(const float* __restrict__ partial, // [16][64][512]
                                const float* __restrict__ gates,   // [64][1536]
                                float* __restrict__ read_out)      // [64][512]
{
    int idx = blockIdx.x * blockDim.x + threadIdx.x; // 32768
    int b = idx >> 9, o = idx & 511;
    float s = 0.f;
    #pragma unroll
    for (int c = 0; c < 16; ++c) s += partial[(c * 64 + b) * 512 + o];
    read_out[idx] = s * gates[b * 1536 + o];
}

extern "C" void kernel_launch(void* const* d_in, const int* in_sizes, int n_in,
                              void* d_out, int out_size, void* d_ws, size_t ws_size,
                              hipStream_t stream) {
    const float* ctrl   = (const float*)d_in[0];
    const float* memory = (const float*)d_in[1];
    const float* obj    = (const float*)d_in[2];
    const float* Wk     = (const float*)d_in[3];
    const float* bk     = (const float*)d_in[4];
    const float* Wsub   = (const float*)d_in[5];
    const float* bsub   = (const float*)d_in[6];
    const float* Wmix   = (const float*)d_in[7];
    const float* bmix   = (const float*)d_in[8];
    const float* Wloc   = (const float*)d_in[9];
    const float* bloc   = (const float*)d_in[10];
    const float* Wgate  = (const float*)d_in[11];
    const float* bgate  = (const float*)d_in[12];
    const float* Wsharp = (const float*)d_in[13];
    const float* bsharp = (const float*)d_in[14];

    float* read_out = (float*)d_out;              // [64][512]
    float* mem_out  = read_out + 64 * 512;        // [64][1024][512]

    float* ws      = (float*)d_ws;
    float* keys    = ws;                 // 64*1536
    float* subs    = keys    + 98304;    // 64*1536
    float* gates   = subs    + 98304;    // 64*1536
    float* locl    = gates   + 98304;    // 64*3072
    float* vk      = locl    + 196608;   // 64*1536
    float* mix     = vk      + 98304;    // 3*64*2
    float* sharp   = mix     + 384;      // 3*64
    float* content = sharp   + 192;      // 3*64*1024
    float* wts     = content + 196608;   // 3*64*1024
    float* partial = wts     + 196608;   // 16*64*512

    // Controller projections (fp32 WMMA GEMMs)
    gemm_ctrl_wmma<<<48, 256, 0, stream>>>(ctrl, Wk,    bk,    keys,  1536, 0);
    gemm_ctrl_wmma<<<48, 256, 0, stream>>>(ctrl, Wsub,  bsub,  subs,  1536, 1);
    gemm_ctrl_wmma<<<48, 256, 0, stream>>>(ctrl, Wgate, bgate, gates, 1536, 1);
    gemm_ctrl_wmma<<<96, 256, 0, stream>>>(ctrl, Wloc,  bloc,  locl,  3072, 0);

    // Small projections + key normalization
    mixsharp_kernel<<<24, 256, 0, stream>>>(ctrl, Wmix, bmix, Wsharp, bsharp, mix, sharp);
    vk_kernel<<<24, 256, 0, stream>>>(keys, subs, vk);

    // Pass 1 over memory: fused row-norm + 3 content dots (b128 loads)
    content_kernel<<<8192, 256, 0, stream>>>(memory, vk, content);

    // Addressing weights
    addr_kernel<<<192, 256, 0, stream>>>(locl, content, mix, sharp, wts);

    // Pass 2 over memory: erase+write+read-partials (b128, NT stores)
    update_kernel<<<512, 256, 0, stream>>>(memory, obj, gates, wts, mem_out, partial);
    finalize_kernel<<<128, 256, 0, stream>>>(partial, gates, read_out);
}